// Boundary_Aggregation_17179869331
// MI455X (gfx1250) — compile-verified
//
#include <hip/hip_runtime.h>

typedef float v2f __attribute__((ext_vector_type(2)));
typedef float v4f __attribute__((ext_vector_type(4)));
typedef float v8f __attribute__((ext_vector_type(8)));

#define Bn 4
#define Cn 32
#define Hn 512
#define Wn 512
#define HW (Hn * Wn)              // 262144
#define KTOT 160                  // 5*C
#define KSTEPS 40                 // 160 / 4

#define VS_BLOCKS ((Bn * Cn * Wn) / 256)   // 256  (vertical: thread per column)
#define HS_BLOCKS ((Bn * Cn * Hn) / 8)     // 8192 (horizontal: wave per row)

// ---------------------------------------------------------------------------
// Merged scan kernel: the vertical (along-H) and horizontal (along-W) running
// maxes are independent (both read x, write disjoint maps), so one launch
// runs both concurrently — the scans are latency-bound, not bandwidth-bound,
// and overlapping them hides the serial-chain latency.
// ---------------------------------------------------------------------------
__global__ __launch_bounds__(256) void scan_kernel(const float* __restrict__ x,
                                                   float* __restrict__ x1,
                                                   float* __restrict__ x2,
                                                   float* __restrict__ x3,
                                                   float* __restrict__ x4) {
    if (blockIdx.x < VS_BLOCKS) {
        // ---- vertical prefix/suffix max along H: one thread per (b,c,w) ----
        int g  = blockIdx.x * 256 + threadIdx.x;   // 0 .. B*C*W-1
        unsigned w  = g & (Wn - 1);
        unsigned bc = g >> 9;
        unsigned base = bc * HW + w;               // 32-bit element offsets

        float run = -INFINITY;
#pragma unroll 8
        for (int h = 0; h < Hn; ++h) {
            unsigned idx = base + (unsigned)h * Wn;
            run = fmaxf(run, x[idx]);
            x1[idx] = run;
        }
        run = -INFINITY;
#pragma unroll 8
        for (int h = Hn - 1; h >= 0; --h) {
            unsigned idx = base + (unsigned)h * Wn;
            run = fmaxf(run, x[idx]);
            x2[idx] = run;
        }
    } else {
        // ---- horizontal prefix/suffix max along W: one wave32 per row ----
        // Lane l owns 4 consecutive floats (b128); per 128-elem chunk:
        // 3-op in-lane scan + 5-step wave scan-max over lane totals + carry.
        int lane = threadIdx.x & 31;
        int wave = threadIdx.x >> 5;
        unsigned row  = (blockIdx.x - VS_BLOCKS) * 8 + wave;  // 0 .. B*C*H-1
        unsigned base = row * Wn;

        // left -> right (prefix max)
        float carry = -INFINITY;
#pragma unroll
        for (int ch = 0; ch < Wn / 128; ++ch) {
            unsigned off = base + ch * 128 + lane * 4;
            v4f v = *(const v4f*)(x + off);
            v.y = fmaxf(v.y, v.x);
            v.z = fmaxf(v.z, v.y);
            v.w = fmaxf(v.w, v.z);
            float incl = v.w;                       // lane total
#pragma unroll
            for (int d = 1; d < 32; d <<= 1) {
                float t = __shfl_up(incl, d, 32);
                if (lane >= d) incl = fmaxf(incl, t);
            }
            float excl = __shfl_up(incl, 1, 32);    // max over lanes 0..lane-1
            float pre  = (lane == 0) ? carry : fmaxf(excl, carry);
            v.x = fmaxf(v.x, pre);
            v.y = fmaxf(v.y, pre);
            v.z = fmaxf(v.z, pre);
            v.w = fmaxf(v.w, pre);
            *(v4f*)(x3 + off) = v;
            carry = fmaxf(carry, __shfl(incl, 31, 32));
        }
        // right -> left (suffix max)
        carry = -INFINITY;
#pragma unroll
        for (int ch = Wn / 128 - 1; ch >= 0; --ch) {
            unsigned off = base + ch * 128 + lane * 4;
            v4f v = *(const v4f*)(x + off);
            v.z = fmaxf(v.z, v.w);
            v.y = fmaxf(v.y, v.z);
            v.x = fmaxf(v.x, v.y);
            float incl = v.x;                       // lane total
#pragma unroll
            for (int d = 1; d < 32; d <<= 1) {
                float t = __shfl_down(incl, d, 32);
                if (lane < 32 - d) incl = fmaxf(incl, t);
            }
            float excl = __shfl_down(incl, 1, 32);  // max over lanes lane+1..31
            float pre  = (lane == 31) ? carry : fmaxf(excl, carry);
            v.x = fmaxf(v.x, pre);
            v.y = fmaxf(v.y, pre);
            v.z = fmaxf(v.z, pre);
            v.w = fmaxf(v.w, pre);
            *(v4f*)(x4 + off) = v;
            carry = fmaxf(carry, __shfl(incl, 0, 32));
        }
    }
}

// ---------------------------------------------------------------------------
// Kernel 2: fused concat + 1x1 conv via V_WMMA_F32_16X16X4_F32.
// Block = 256 threads (8 waves); each block owns 128 consecutive w of one
// (b,h) row; each wave owns a 16-pixel M-tile. K = 160 (5 sources x 32 ch),
// stepped by 4. N = 32 out channels = two 16-wide N-tiles sharing A.
//
// Outer loop over the 5 concat sources is unrolled so each source pointer is
// a direct kernarg pointer (global address space statically known ->
// global_load, not flat_load), with 32-bit element offsets.
//
// A layout (16x4 f32, ISA 7.12.2): lane m = pixel; lanes 0-15 carry K={k0,k0+1},
// lanes 16-31 carry K={k0+2,k0+3}.  B (4x16) mirrored: per-lane float2 of
// W[n][k], pre-swizzled into LDS.  D (16x16 f32, 8 VGPRs): lanes 0-15 M=r,
// lanes 16-31 M=8+r, N=lane%16 -> transposed through LDS for coalesced stores.
// ---------------------------------------------------------------------------
__global__ __launch_bounds__(256) void conv1x1_wmma_kernel(
        const float* __restrict__ x,  const float* __restrict__ x1,
        const float* __restrict__ x2, const float* __restrict__ x3,
        const float* __restrict__ x4, const float* __restrict__ Wmat,
        const float* __restrict__ bias, float* __restrict__ out) {
    __shared__ v2f   sWB[KSTEPS * 2 * 32];     // [step][ntile][lane], 20 KB
    __shared__ float sOut[Cn][128 + 4];        // D transpose staging, ~17 KB

    const int tid = threadIdx.x;

    // Stage pre-swizzled B operands: B[k][n] = Wmat[n*160 + k]
    for (int e = tid; e < KSTEPS * 64; e += 256) {
        int s = e >> 6;             // k-step
        int l = e & 31;             // lane
        int t = (e >> 5) & 1;       // n-tile
        int n = (l & 15) + 16 * t;
        int k = s * 4 + 2 * (l >> 4);
        v2f wv;
        wv.x = Wmat[n * KTOT + k];
        wv.y = Wmat[n * KTOT + k + 1];
        sWB[e] = wv;
    }
    __syncthreads();

    const int lane = tid & 31;
    const int wave = tid >> 5;
    const int m    = lane & 15;     // pixel within M-tile (A layout)
    const int half = lane >> 4;     // K half-select (A layout)

    int blk    = blockIdx.x;        // B*H*(W/128) = 8192
    int wblk   = blk & 3;
    int bh     = blk >> 2;
    int b      = bh >> 9;           // / Hn
    int h      = bh & (Hn - 1);
    int wblock = wblk * 128;
    int wbase  = wblock + 16 * wave;

    // 32-bit element offset of this lane's pixel at channel (2*half) of any map
    const unsigned pixbase =
        (unsigned)b * (Cn * HW) + (unsigned)h * Wn + (unsigned)(wbase + m)
        + (unsigned)(2 * half) * HW;

    v8f acc0 = {};
    v8f acc1 = {};

#pragma unroll
    for (int g = 0; g < 5; ++g) {
        // statically selected kernarg pointer -> global address space
        const float* __restrict__ src =
            (g == 0) ? x : (g == 1) ? x1 : (g == 2) ? x2 : (g == 3) ? x3 : x4;
#pragma unroll 4
        for (int s8 = 0; s8 < 8; ++s8) {
            const int s = g * 8 + s8;
            unsigned idx = pixbase + (unsigned)(s8 * 4) * HW;
            v2f a;
            a.x = src[idx];          // K = k0 + 2*half
            a.y = src[idx + HW];     // K = k0 + 2*half + 1
            v2f b0 = sWB[s * 64 + lane];
            v2f b1 = sWB[s * 64 + 32 + lane];
            acc0 = __builtin_amdgcn_wmma_f32_16x16x4_f32(
                false, a, false, b0, (short)0, acc0, false, false);
            acc1 = __builtin_amdgcn_wmma_f32_16x16x4_f32(
                false, a, false, b1, (short)0, acc1, false, false);
        }
    }

    // D -> LDS transpose: lane l, VGPR r holds (pixel = r + 8*(l>>4), o = l%16 + 16*tile)
    int n0 = lane & 15;
    int mbase = 8 * half;
#pragma unroll
    for (int r = 0; r < 8; ++r) {
        int mm = mbase + r;
        sOut[n0][16 * wave + mm]      = acc0[r];
        sOut[n0 + 16][16 * wave + mm] = acc1[r];
    }
    __syncthreads();

    // Coalesced store: 128 consecutive w per out-channel row, bias fused.
    unsigned outbase = (unsigned)b * (Cn * HW) + (unsigned)h * Wn + wblock;
    for (int e = tid; e < Cn * 128; e += 256) {
        int o  = e >> 7;
        int wl = e & 127;
        out[outbase + (unsigned)o * HW + wl] = sOut[o][wl] + bias[o];
    }
}

// ---------------------------------------------------------------------------
extern "C" void kernel_launch(void* const* d_in, const int* in_sizes, int n_in,
                              void* d_out, int out_size, void* d_ws, size_t ws_size,
                              hipStream_t stream) {
    const float* x    = (const float*)d_in[0];   // [4,32,512,512]
    const float* Wm   = (const float*)d_in[1];   // [32,160,1,1]
    const float* bias = (const float*)d_in[2];   // [32]
    float*       out  = (float*)d_out;           // [4,32,512,512]

    const size_t N = (size_t)Bn * Cn * HW;       // elements per map
    float* ws1 = (float*)d_ws;                   // needs 4*N*4 = 512 MiB ws
    float* ws2 = ws1 + N;
    float* ws3 = ws2 + N;
    float* ws4 = ws3 + N;

    // 1) all four directional scans in one launch (independent work items)
    scan_kernel<<<VS_BLOCKS + HS_BLOCKS, 256, 0, stream>>>(x, ws1, ws2, ws3, ws4);
    // 2) fused concat + 1x1 conv (WMMA f32): B*H*(W/128) = 8192 blocks
    conv1x1_wmma_kernel<<<Bn * Hn * (Wn / 128), 256, 0, stream>>>(
        x, ws1, ws2, ws3, ws4, Wm, bias, out);
}